// MultiReDiffusion_46402826666245
// MI455X (gfx1250) — compile-verified
//
#include <hip/hip_runtime.h>

typedef __attribute__((ext_vector_type(2))) float v2f;
typedef __attribute__((ext_vector_type(8))) float v8f;

#define N_DIM 2048
#define R_DIM 4
#define S_DIM 6
#define D_DIM 32
#define ROWS_PER_BLOCK 32
#define LDS_STRIDE 34   // padded row stride in floats: even (8B-aligned b64) + bank-spread

__global__ __launch_bounds__(256) void MultiReDiffusion_fused_kernel(
    const float* __restrict__ theta, const float* __restrict__ tpar,
    const float* __restrict__ A,     const float* __restrict__ X,
    const float* __restrict__ fcw,   const float* __restrict__ fcb,
    const float* __restrict__ convw, const float* __restrict__ convb,
    const float* __restrict__ p0,    const float* __restrict__ p1,
    float* __restrict__ out)
{
    __shared__ float lds_diff[R_DIM * ROWS_PER_BLOCK * LDS_STRIDE];
    __shared__ float lds_h  [R_DIM * ROWS_PER_BLOCK * LDS_STRIDE];

    const int tid   = threadIdx.x;
    const int lane  = tid & 31;
    const int wid   = tid >> 5;
    const int r     = wid & 3;       // relation handled by this wave
    const int sub   = wid >> 2;      // row half-tile 0/1
    const int b     = blockIdx.y;
    const int n0    = blockIdx.x * ROWS_PER_BLOCK;

    const int lhalf = lane >> 4;     // 0: lanes 0-15, 1: lanes 16-31
    const int llow  = lane & 15;

    // coef[r] = sum_s theta[r,s] * t[r,s]  (uniform per wave, 6 FMAs)
    float coef = 0.f;
    #pragma unroll
    for (int s = 0; s < S_DIM; ++s)
        coef += theta[r * S_DIM + s] * tpar[r * S_DIM + s];

    // ---------------- Stage 1: diff = coef * (A[b,r] @ X[b,r]) ----------------
    // Wave computes rows [n0 + sub*16, +16), all 32 columns, K = 2048.
    const long br = (long)(b * R_DIM + r);
    // A fragment base: row (M) in lanes, K/K+1 per lane, K+2/K+3 in upper half-wave
    const float* Abase = A + (br * N_DIM + (long)(n0 + sub * 16 + llow)) * N_DIM + 2 * lhalf;
    // B fragment base: X[b,r, k + 2*lhalf, llow]
    const float* Xbase = X + br * (long)N_DIM * D_DIM + (2 * lhalf) * D_DIM + llow;

    v8f acc0 = {}; v8f acc1 = {};
    #pragma unroll 4
    for (int k = 0; k < N_DIM; k += 4) {
        // stream A non-temporally: read-once data, keep L2 for X
        v2f afrag = __builtin_nontemporal_load((const v2f*)(Abase + k));
        const float* xp = Xbase + k * D_DIM;
        v2f b0, b1;
        b0.x = xp[0];              b0.y = xp[D_DIM];        // cols 0..15
        b1.x = xp[16];             b1.y = xp[D_DIM + 16];   // cols 16..31
        acc0 = __builtin_amdgcn_wmma_f32_16x16x4_f32(false, afrag, false, b0,
                                                     (short)0, acc0, false, false);
        acc1 = __builtin_amdgcn_wmma_f32_16x16x4_f32(false, afrag, false, b1,
                                                     (short)0, acc1, false, false);
    }
    acc0 = acc0 * coef;
    acc1 = acc1 * coef;

    // C layout -> LDS: VGPR j holds M=j (lanes 0-15) / M=j+8 (lanes 16-31), N=llow
    {
        float* base = lds_diff + (r * ROWS_PER_BLOCK + sub * 16) * LDS_STRIDE;
        #pragma unroll
        for (int j = 0; j < 8; ++j) {
            const int m = j + 8 * lhalf;
            base[m * LDS_STRIDE + llow]      = acc0[j];
            base[m * LDS_STRIDE + 16 + llow] = acc1[j];
        }
    }
    __syncthreads();

    // ---------------- Stage 2: fc = diff @ fc_w[r] + fc_b[r], PReLU0 ----------
    v8f f0 = {}; v8f f1 = {};
    const float* wbase = fcw + r * D_DIM * D_DIM + (2 * lhalf) * D_DIM + llow;
    const float* dbase = lds_diff + (r * ROWS_PER_BLOCK + sub * 16 + llow) * LDS_STRIDE
                                  + 2 * lhalf;
    #pragma unroll
    for (int k = 0; k < D_DIM; k += 4) {
        v2f afrag = *(const v2f*)(dbase + k);     // ds_load_b64, 8B aligned
        const float* wp = wbase + k * D_DIM;
        v2f b0, b1;
        b0.x = wp[0];   b0.y = wp[D_DIM];
        b1.x = wp[16];  b1.y = wp[D_DIM + 16];
        f0 = __builtin_amdgcn_wmma_f32_16x16x4_f32(false, afrag, false, b0,
                                                   (short)0, f0, false, false);
        f1 = __builtin_amdgcn_wmma_f32_16x16x4_f32(false, afrag, false, b1,
                                                   (short)0, f1, false, false);
    }

    const float slope0 = p0[0];
    const float bias0  = fcb[r * D_DIM + llow];
    const float bias1  = fcb[r * D_DIM + 16 + llow];
    {
        float* hbase = lds_h + (r * ROWS_PER_BLOCK + sub * 16) * LDS_STRIDE;
        #pragma unroll
        for (int j = 0; j < 8; ++j) {
            const int m = j + 8 * lhalf;
            float v0 = f0[j] + bias0;
            float v1 = f1[j] + bias1;
            v0 = (v0 >= 0.f) ? v0 : slope0 * v0;
            v1 = (v1 >= 0.f) ? v1 : slope0 * v1;
            hbase[m * LDS_STRIDE + llow]      = v0;
            hbase[m * LDS_STRIDE + 16 + llow] = v1;
        }
    }
    __syncthreads();

    // ---------------- Stage 3: relation mixing (1x1 conv) + PReLU1 ------------
    // mixed[q,n,e] = sum_r conv_w[q,r] * h[r,n,e] + conv_b[q]
    const float slope1 = p1[0];
    #pragma unroll
    for (int i = 0; i < 16; ++i) {
        const int idx = i * 256 + tid;         // [0, 4096): q(2) | n(5) | e(5)
        const int q   = idx >> 10;
        const int ne  = idx & 1023;
        const int n   = ne >> 5;
        const int e   = ne & 31;
        float acc = convb[q];
        #pragma unroll
        for (int rr = 0; rr < R_DIM; ++rr)
            acc += convw[q * R_DIM + rr] *
                   lds_h[(rr * ROWS_PER_BLOCK + n) * LDS_STRIDE + e];
        acc = (acc >= 0.f) ? acc : slope1 * acc;
        out[(((long)b * R_DIM + q) * N_DIM + (n0 + n)) * D_DIM + e] = acc;
    }
}

extern "C" void kernel_launch(void* const* d_in, const int* in_sizes, int n_in,
                              void* d_out, int out_size, void* d_ws, size_t ws_size,
                              hipStream_t stream) {
    const float* theta = (const float*)d_in[0];
    const float* tpar  = (const float*)d_in[1];
    const float* A     = (const float*)d_in[2];
    const float* X     = (const float*)d_in[3];
    const float* fcw   = (const float*)d_in[4];
    const float* fcb   = (const float*)d_in[5];
    const float* convw = (const float*)d_in[6];
    const float* convb = (const float*)d_in[7];
    const float* p0    = (const float*)d_in[8];
    const float* p1    = (const float*)d_in[9];
    float* out = (float*)d_out;

    dim3 grid(N_DIM / ROWS_PER_BLOCK, 8);   // 64 row-tiles x B=8
    dim3 block(256);                        // 8 waves: 4 relations x 2 row halves
    MultiReDiffusion_fused_kernel<<<grid, block, 0, stream>>>(
        theta, tpar, A, X, fcw, fcb, convw, convb, p0, p1, out);
}